// PointConvMiniSqueeze_9354438770950
// MI455X (gfx1250) — compile-verified
//
#include <hip/hip_runtime.h>
#include <hip/hip_bf16.h>

#define BB 8
#define CC 3
#define NN 4096
#define NS 10

typedef __attribute__((ext_vector_type(2))) float v2f;
typedef __attribute__((ext_vector_type(8))) float v8f;

// ---------------------------------------------------------------------------
// Kernel 0: squared norms sq[b][n] = sum_c pos[b][c][n]^2
// ---------------------------------------------------------------------------
__global__ __launch_bounds__(256) void sq_kernel(const float* __restrict__ pos,
                                                 float* __restrict__ sq) {
  int i = blockIdx.x * 256 + threadIdx.x;      // over B*N
  int b = i >> 12;
  int n = i & (NN - 1);
  const float* pb = pos + (size_t)b * CC * NN + n;
  float x = pb[0], y = pb[NN], z = pb[2 * NN];
  sq[i] = x * x + y * y + z * z;
}

// ---------------------------------------------------------------------------
// Kernel 1: fuse the two linear layers: Wc[p][k] = sum_o w2[p][o]*w1f[o][k],
//           bc[p] = b2[p] + sum_o w2[p][o]*b1[o].   (w1 is already (64,48) flat)
// ---------------------------------------------------------------------------
__global__ __launch_bounds__(256) void wc_kernel(const float* __restrict__ w1,
                                                 const float* __restrict__ b1,
                                                 const float* __restrict__ w2,
                                                 const float* __restrict__ b2,
                                                 float* __restrict__ Wc,
                                                 float* __restrict__ bc) {
  int t = threadIdx.x;
  for (int e = t; e < 64 * 48; e += 256) {
    int p = e / 48, k = e % 48;
    float s = 0.f;
    for (int o = 0; o < 64; ++o) s += w2[p * 64 + o] * w1[o * 48 + k];
    Wc[e] = s;
  }
  if (t < 64) {
    float s = b2[t];
    for (int o = 0; o < 64; ++o) s += w2[t * 64 + o] * b1[o];
    bc[t] = s;
  }
}

// ---------------------------------------------------------------------------
// Kernel 2: kNN top-10 per point. One wave per query row; whole batch's
// points + norms staged in LDS. Key = 2*dot - sq[m]  (== -d2 + const).
// Tie-break: smaller index first (matches jax.lax.top_k).
// ---------------------------------------------------------------------------
__global__ __launch_bounds__(256) void knn_kernel(const float* __restrict__ pos,
                                                  const float* __restrict__ sq,
                                                  int* __restrict__ idx) {
  extern __shared__ float smem[];
  float* lx  = smem;
  float* ly  = smem + NN;
  float* lz  = smem + 2 * NN;
  float* lsq = smem + 3 * NN;
  float* mv  = smem + 4 * NN;            // 8*320 floats
  int*   mi  = (int*)(smem + 4 * NN + 8 * 320);

  int b  = blockIdx.x >> 9;              // 512 blocks per batch
  int n0 = (blockIdx.x & 511) * 8;
  const float* pb = pos + (size_t)b * CC * NN;

  for (int i = threadIdx.x; i < NN; i += 256) {
    lx[i]  = pb[i];
    ly[i]  = pb[NN + i];
    lz[i]  = pb[2 * NN + i];
    lsq[i] = sq[b * NN + i];
  }
  __syncthreads();

  int wave = threadIdx.x >> 5;
  int lane = threadIdx.x & 31;
  int n = n0 + wave;
  float xn = lx[n], yn = ly[n], zn = lz[n];

  float tv[NS];
  int   ti[NS];
#pragma unroll
  for (int j = 0; j < NS; ++j) { tv[j] = -3.4e38f; ti[j] = 0x7fffffff; }

  for (int m = lane; m < NN; m += 32) {
    float dot = xn * lx[m] + yn * ly[m] + zn * lz[m];
    float key = 2.0f * dot - lsq[m];
    if (key > tv[NS - 1] || (key == tv[NS - 1] && m < ti[NS - 1])) {
      tv[NS - 1] = key; ti[NS - 1] = m;
#pragma unroll
      for (int j = NS - 1; j > 0; --j) {
        bool sw = (tv[j] > tv[j - 1]) || (tv[j] == tv[j - 1] && ti[j] < ti[j - 1]);
        if (sw) {
          float fv = tv[j]; tv[j] = tv[j - 1]; tv[j - 1] = fv;
          int   iv = ti[j]; ti[j] = ti[j - 1]; ti[j - 1] = iv;
        }
      }
    }
  }

#pragma unroll
  for (int j = 0; j < NS; ++j) {
    mv[wave * 320 + lane * NS + j] = tv[j];
    mi[wave * 320 + lane * NS + j] = ti[j];
  }
  __syncthreads();

  if (lane == 0) {
    int* op = idx + ((size_t)b * NN + n) * NS;
    for (int r = 0; r < NS; ++r) {
      float best = -3.4e38f; int bi = 0x7fffffff; int bp = 0;
      for (int t = 0; t < 320; ++t) {
        float v = mv[wave * 320 + t];
        int   ii = mi[wave * 320 + t];
        if (v > best || (v == best && ii < bi)) { best = v; bi = ii; bp = t; }
      }
      op[r] = bi;
      mv[wave * 320 + bp] = -3.4e38f;
      mi[wave * 320 + bp] = 0x7fffffff;
    }
  }
}

// ---------------------------------------------------------------------------
// Compile-time mapping of feature index k (0..47) -> (channel, sample) of the
// squeezed 2x2 tensor: k = cc*4 + hh*2 + ww, cc = block*3 + c,
// blocks [a,d,b,c] of the 0::2/1::2 picks, sample via INDS on the 4x4 grid.
// ---------------------------------------------------------------------------
__device__ constexpr int kc_of(int k) { return (k >> 2) % 3; }

__device__ constexpr int ks_of(int k) {
  constexpr int INDS[16] = {0, 1, 2, 3, 2, 3, 4, 5, 4, 5, 6, 7, 6, 7, 8, 9};
  int cc = k >> 2, r = k & 3, hh = r >> 1, ww = r & 1;
  int blk = cc / 3;
  int h = (blk == 0 || blk == 2) ? 2 * hh : 2 * hh + 1;
  int w = (blk == 0) ? 2 * ww : (blk == 1) ? 2 * ww + 1
                     : (blk == 2) ? 2 * ww + 1 : 2 * ww;
  return INDS[h * 4 + w];
}

// One K=4 WMMA step for a literal KB. constexpr locals force frontend
// constant evaluation, so g[][] subscripts are literals (pure register refs).
#define WMMA_STEP(KB)                                                          \
  {                                                                            \
    constexpr int c0 = kc_of((KB) + 0), s0 = ks_of((KB) + 0);                  \
    constexpr int c1 = kc_of((KB) + 1), s1 = ks_of((KB) + 1);                  \
    constexpr int c2 = kc_of((KB) + 2), s2 = ks_of((KB) + 2);                  \
    constexpr int c3 = kc_of((KB) + 3), s3 = ks_of((KB) + 3);                  \
    v2f a;                                                                     \
    a.x = hi ? g[c2][s2] : g[c0][s0];                                          \
    a.y = hi ? g[c3][s3] : g[c1][s1];                                          \
    const v2f* w0 = (const v2f*)(wrow0 + (KB));                                \
    const v2f* w1 = (const v2f*)(wrow1 + (KB));                                \
    const v2f* w2 = (const v2f*)(wrow2 + (KB));                                \
    const v2f* w3 = (const v2f*)(wrow3 + (KB));                                \
    acc0 = __builtin_amdgcn_wmma_f32_16x16x4_f32(false, a, false, *w0,         \
                                                 (short)0, acc0, false, false);\
    acc1 = __builtin_amdgcn_wmma_f32_16x16x4_f32(false, a, false, *w1,         \
                                                 (short)0, acc1, false, false);\
    acc2 = __builtin_amdgcn_wmma_f32_16x16x4_f32(false, a, false, *w2,         \
                                                 (short)0, acc2, false, false);\
    acc3 = __builtin_amdgcn_wmma_f32_16x16x4_f32(false, a, false, *w3,         \
                                                 (short)0, acc3, false, false);\
  }

// ---------------------------------------------------------------------------
// Kernel 3: gather + fused MLP via V_WMMA_F32_16X16X4_F32.
// GEMM: out[b][p][n] = sum_k x[b,n][k] * Wc[p][k] + bc[p]
//   M = 16 points per wave-tile, N = 64 (4 tiles of 16), K = 48 (12 x K=4).
// ---------------------------------------------------------------------------
__global__ __launch_bounds__(128) void mlp_wmma_kernel(const float* __restrict__ pos,
                                                       const int* __restrict__ idx,
                                                       const float* __restrict__ Wc,
                                                       const float* __restrict__ bc,
                                                       float* __restrict__ out) {
  int wave = threadIdx.x >> 5;
  int lane = threadIdx.x & 31;
  int tile = blockIdx.x * 4 + wave;      // over (B*N)/16 = 2048 tiles
  int row16 = tile * 16;                 // global flattened point base (b*N + n0)
  int b  = row16 >> 12;
  int n0 = row16 & (NN - 1);
  int mrow = lane & 15;                  // A-matrix row within tile
  bool hi  = (lane >> 4) != 0;           // false: K={0,1}, true: K={2,3}

  // gather the 10 neighbor values for this lane's row, for each channel
  int gi[NS];
  const int* ip = idx + (size_t)(row16 + mrow) * NS;
#pragma unroll
  for (int s = 0; s < NS; ++s) gi[s] = ip[s];

  float g[CC][NS];
  const float* pb = pos + (size_t)b * CC * NN;
#pragma unroll
  for (int c = 0; c < CC; ++c)
#pragma unroll
    for (int s = 0; s < NS; ++s) g[c][s] = pb[c * NN + gi[s]];

  v8f acc0 = {0, 0, 0, 0, 0, 0, 0, 0};
  v8f acc1 = {0, 0, 0, 0, 0, 0, 0, 0};
  v8f acc2 = {0, 0, 0, 0, 0, 0, 0, 0};
  v8f acc3 = {0, 0, 0, 0, 0, 0, 0, 0};

  int p0 = lane & 15;
  int koff = hi ? 2 : 0;                 // K sub-offset within the K=4 step
  // B fragment row bases: B[k][p] = Wc[p][k], half-split along K
  const float* wrow0 = Wc + (size_t)(0 * 16 + p0) * 48 + koff;
  const float* wrow1 = Wc + (size_t)(1 * 16 + p0) * 48 + koff;
  const float* wrow2 = Wc + (size_t)(2 * 16 + p0) * 48 + koff;
  const float* wrow3 = Wc + (size_t)(3 * 16 + p0) * 48 + koff;

  WMMA_STEP(0)  WMMA_STEP(4)  WMMA_STEP(8)  WMMA_STEP(12)
  WMMA_STEP(16) WMMA_STEP(20) WMMA_STEP(24) WMMA_STEP(28)
  WMMA_STEP(32) WMMA_STEP(36) WMMA_STEP(40) WMMA_STEP(44)

  // C/D layout: VGPR j -> row (j + half*8), col (lane&15). out is (B, 64, N).
  int nbase = n0 + (hi ? 8 : 0);
#pragma unroll
  for (int t = 0; t < 4; ++t) {
    v8f acc = (t == 0) ? acc0 : (t == 1) ? acc1 : (t == 2) ? acc2 : acc3;
    int p = t * 16 + p0;
    float bias = bc[p];
    float* op = out + ((size_t)b * 64 + p) * NN + nbase;
#pragma unroll
    for (int j = 0; j < 8; ++j) op[j] = acc[j] + bias;
  }
}

// ---------------------------------------------------------------------------
extern "C" void kernel_launch(void* const* d_in, const int* in_sizes, int n_in,
                              void* d_out, int out_size, void* d_ws, size_t ws_size,
                              hipStream_t stream) {
  const float* pos = (const float*)d_in[0];
  const float* w1  = (const float*)d_in[1];
  const float* b1  = (const float*)d_in[2];
  const float* w2  = (const float*)d_in[3];
  const float* b2  = (const float*)d_in[4];
  float* out = (float*)d_out;

  float* sq  = (float*)d_ws;                 // B*N = 32768 floats
  int*   idx = (int*)(sq + BB * NN);         // B*N*10 ints
  float* Wc  = (float*)(idx + BB * NN * NS); // 64*48 floats
  float* bc  = Wc + 64 * 48;                 // 64 floats

  sq_kernel<<<(BB * NN) / 256, 256, 0, stream>>>(pos, sq);
  wc_kernel<<<1, 256, 0, stream>>>(w1, b1, w2, b2, Wc, bc);

  // 86016 bytes dynamic LDS: 4*4096 staged floats + 8*320 merge (val+idx)
  knn_kernel<<<BB * (NN / 8), 256, 86016, stream>>>(pos, sq, idx);

  mlp_wmma_kernel<<<(BB * NN / 16) / 4, 128, 0, stream>>>(pos, idx, Wc, bc, out);
}